// GATStack_7301444403634
// MI455X (gfx1250) — compile-verified
//
#include <hip/hip_runtime.h>
#include <hip/hip_bf16.h>

// ---------------------------------------------------------------------------
// GATv2 stack (N=50000 nodes, E=640000 edges, IN=D=128, H=4 heads, C=32, L=3)
// MI455X / gfx1250, wave32.  GEMMs use V_WMMA_F32_16X16X4_F32 (fp32 matrix
// path) with LDS-staged weights + activations (b128 staging, ds_load_2addr
// fragment reads).  Edge phase is bandwidth/atomic bound -> wave-per-edge
// float4 I/O, native f32 global atomics, int-keyed atomic max for softmax.
// ---------------------------------------------------------------------------

#define GN  50000
#define GE  640000
#define GIN 128
#define GH  4
#define GC  32
#define GD  128
#define GL  3
#define NEG_SLOPE 0.2f
#define LN_EPS    1e-5f

#define BT    4     // 16-row tiles per block (64 rows)
#define ASTR  132   // A LDS row stride (floats): 132%64=4 -> conflict-free
#define WSTR  136   // W LDS k-stride (floats): 2*136%64=16 -> lane halves
                    // land in disjoint 16-bank windows

typedef float v2f __attribute__((ext_vector_type(2)));
typedef float v8f __attribute__((ext_vector_type(8)));

// monotonic float <-> signed-int key (for atomicMax-based segment max)
__device__ __forceinline__ int f32_to_ordered(float f) {
    int i = __float_as_int(f);
    return (i >= 0) ? i : (i ^ 0x7fffffff);
}
__device__ __forceinline__ float ordered_to_f32(int s) {
    return __int_as_float((s >= 0) ? s : (s ^ 0x7fffffff));
}

__device__ __forceinline__ float wave32_sum(float v) {
    #pragma unroll
    for (int off = 1; off < 32; off <<= 1) v += __shfl_xor(v, off, 32);
    return v;
}

// ------------------------------- fill --------------------------------------
__global__ __launch_bounds__(256) void fill_u32(unsigned* __restrict__ p,
                                                unsigned v, int n) {
    int i = blockIdx.x * 256 + threadIdx.x;
    if (i < n) p[i] = v;
}

// --------------------- fused dual GEMM: xl = xWl+bl, xr = xWr+br -----------
// Block: 256 threads / 8 waves, 64 rows (4 row tiles).  Both 128x128 weight
// matrices staged in LDS once per block; wave w owns output columns
// [16w,16w+16) of BOTH xl and xr across all 4 row tiles -> 8 independent
// WMMA accumulator chains per wave, B fragments reused 4x, A fragments 2x.
__global__ __launch_bounds__(256)
void gemm_xl_xr(const float* __restrict__ x,
                const float* __restrict__ Wl, const float* __restrict__ bl,
                const float* __restrict__ Wr, const float* __restrict__ br,
                float* __restrict__ xl, float* __restrict__ xr)
{
    __shared__ float WldsL[GIN * WSTR];       // Wl[k][n], stride-padded
    __shared__ float WldsR[GIN * WSTR];       // Wr[k][n]
    __shared__ float As[BT * 16 * ASTR];      // 64 x 128 activation tile

    const int row0 = blockIdx.x * (BT * 16);
    const int tid  = threadIdx.x;

    // stage weights: 128x128 per matrix, float4 granules (16 iters/thread)
    #pragma unroll
    for (int i = tid; i < GIN * (GD / 4); i += 256) {
        const int k = i >> 5;
        const int j = (i & 31) << 2;
        *(float4*)&WldsL[k * WSTR + j] = *(const float4*)(Wl + (size_t)k * GD + j);
        *(float4*)&WldsR[k * WSTR + j] = *(const float4*)(Wr + (size_t)k * GD + j);
    }
    // stage 64x128 A tile, zero-padded past N (8 iters/thread)
    #pragma unroll
    for (int i = tid; i < BT * 16 * (GIN / 4); i += 256) {
        const int r = i >> 5;
        const int j = (i & 31) << 2;
        const int grow = row0 + r;
        float4 v = make_float4(0.f, 0.f, 0.f, 0.f);
        if (grow < GN) v = *(const float4*)(x + (size_t)grow * GIN + j);
        *(float4*)&As[r * ASTR + j] = v;
    }
    __syncthreads();

    const int wave  = tid >> 5;          // column tile 0..7
    const int lane  = tid & 31;
    const int m     = lane & 15;         // A row within tile
    const int khalf = lane >> 4;         // 0: K=k0,k0+1  1: K=k0+2,k0+3
    const int ncol  = (wave << 4) + (lane & 15);

    v8f accL[BT] = {{0}};
    v8f accR[BT] = {{0}};

    #pragma unroll 8
    for (int kt = 0; kt < GIN; kt += 4) {
        const int k0 = kt + (khalf << 1);
        v2f bL, bR;
        bL.x = WldsL[k0 * WSTR + ncol];
        bL.y = WldsL[(k0 + 1) * WSTR + ncol];
        bR.x = WldsR[k0 * WSTR + ncol];
        bR.y = WldsR[(k0 + 1) * WSTR + ncol];
        #pragma unroll
        for (int t = 0; t < BT; ++t) {
            v2f a;
            a.x = As[(t * 16 + m) * ASTR + k0];
            a.y = As[(t * 16 + m) * ASTR + k0 + 1];
            accL[t] = __builtin_amdgcn_wmma_f32_16x16x4_f32(
                false, a, false, bL, (short)0, accL[t], false, false);
            accR[t] = __builtin_amdgcn_wmma_f32_16x16x4_f32(
                false, a, false, bR, (short)0, accR[t], false, false);
        }
    }

    // C/D layout: VGPR r -> row = r + 8*(lane>=16), col = 16*wave + (lane&15)
    const int col = (wave << 4) + (lane & 15);
    const float bLc = bl[col], bRc = br[col];
    #pragma unroll
    for (int t = 0; t < BT; ++t) {
        #pragma unroll
        for (int r = 0; r < 8; ++r) {
            const int grow = row0 + t * 16 + r + (khalf << 3);
            if (grow < GN) {
                xl[(size_t)grow * GD + col] = accL[t][r] + bLc;
                xr[(size_t)grow * GD + col] = accR[t][r] + bRc;
            }
        }
    }
}

// ---------------- edge logits + segment max (wave per edge) ----------------
__global__ __launch_bounds__(256)
void edge_logit(const float* __restrict__ xl, const float* __restrict__ xr,
                const int* __restrict__ srcA, const int* __restrict__ dstA,
                const float* __restrict__ att,
                float* __restrict__ logit, int* __restrict__ mOrd)
{
    const int e = (blockIdx.x << 3) + (threadIdx.x >> 5);
    if (e >= GE) return;
    const int lane = threadIdx.x & 31;
    const int s = srcA[e], d = dstA[e];

    const float4 av = ((const float4*)(xl + (size_t)s * GD))[lane];
    const float4 bv = ((const float4*)(xr + (size_t)d * GD))[lane];
    float4 t;
    t.x = av.x + bv.x; t.y = av.y + bv.y; t.z = av.z + bv.z; t.w = av.w + bv.w;
    t.x = (t.x > 0.f) ? t.x : NEG_SLOPE * t.x;
    t.y = (t.y > 0.f) ? t.y : NEG_SLOPE * t.y;
    t.z = (t.z > 0.f) ? t.z : NEG_SLOPE * t.z;
    t.w = (t.w > 0.f) ? t.w : NEG_SLOPE * t.w;

    const float4 at = ((const float4*)att)[lane];
    float p = t.x * at.x + t.y * at.y + t.z * at.z + t.w * at.w;
    // reduce within the 8 lanes of each head (lanes 8h..8h+7)
    p += __shfl_xor(p, 1, 32);
    p += __shfl_xor(p, 2, 32);
    p += __shfl_xor(p, 4, 32);

    if ((lane & 7) == 0) {
        const int h = lane >> 3;
        logit[(size_t)e * GH + h] = p;
        atomicMax(&mOrd[d * GH + h], f32_to_ordered(p));
    }
}

// ------------------- exp(logit - m[dst]) + segment sum ---------------------
__global__ __launch_bounds__(256)
void edge_exp(float* __restrict__ logit_a,   // in: logit, out: unnormalized a
              const int* __restrict__ dstA, const int* __restrict__ mOrd,
              float* __restrict__ denom)
{
    const int tid = blockIdx.x * 256 + threadIdx.x;
    if (tid >= GE * GH) return;
    const int e = tid >> 2, h = tid & 3;
    const int d = dstA[e];
    const float mf = ordered_to_f32(mOrd[d * GH + h]);
    const float av = __expf(logit_a[tid] - mf);
    logit_a[tid] = av;
    unsafeAtomicAdd(&denom[d * GH + h], av);
}

// --------------- weighted scatter aggregate (wave per edge) ----------------
__global__ __launch_bounds__(256)
void edge_aggr(const float* __restrict__ aArr, const float* __restrict__ denom,
               const float* __restrict__ xl,
               const int* __restrict__ srcA, const int* __restrict__ dstA,
               float* __restrict__ agg)
{
    const int e = (blockIdx.x << 3) + (threadIdx.x >> 5);
    if (e >= GE) return;
    const int lane = threadIdx.x & 31;
    const int s = srcA[e], d = dstA[e];
    const int h = lane >> 3;
    const float w = aArr[(size_t)e * GH + h] / (denom[d * GH + h] + 1e-16f);
    const float4 xv = ((const float4*)(xl + (size_t)s * GD))[lane];
    float* out = agg + (size_t)d * GD + (lane << 2);
    unsafeAtomicAdd(out + 0, w * xv.x);
    unsafeAtomicAdd(out + 1, w * xv.y);
    unsafeAtomicAdd(out + 2, w * xv.z);
    unsafeAtomicAdd(out + 3, w * xv.w);
}

// --------- per-node: +bias, ReLU, LayerNorm, residual (wave per node) ------
__global__ __launch_bounds__(256)
void node_fin(const float* __restrict__ agg, const float* __restrict__ bias,
              const float* __restrict__ gamma, const float* __restrict__ beta,
              float* __restrict__ xio)
{
    const int n = (blockIdx.x << 3) + (threadIdx.x >> 5);
    if (n >= GN) return;
    const int lane = threadIdx.x & 31;

    const float4 a4 = ((const float4*)(agg + (size_t)n * GD))[lane];
    const float4 b4 = ((const float4*)bias)[lane];
    float4 hv;
    hv.x = fmaxf(a4.x + b4.x, 0.f);
    hv.y = fmaxf(a4.y + b4.y, 0.f);
    hv.z = fmaxf(a4.z + b4.z, 0.f);
    hv.w = fmaxf(a4.w + b4.w, 0.f);

    const float mu = wave32_sum(hv.x + hv.y + hv.z + hv.w) * (1.0f / GD);
    float4 c;
    c.x = hv.x - mu; c.y = hv.y - mu; c.z = hv.z - mu; c.w = hv.w - mu;
    const float var = wave32_sum(c.x * c.x + c.y * c.y + c.z * c.z + c.w * c.w)
                      * (1.0f / GD);
    const float inv = rsqrtf(var + LN_EPS);

    const float4 g4 = ((const float4*)gamma)[lane];
    const float4 e4 = ((const float4*)beta)[lane];
    float4 r4 = ((float4*)(xio + (size_t)n * GD))[lane];
    float4 o;
    o.x = c.x * inv * g4.x + e4.x + r4.x;
    o.y = c.y * inv * g4.y + e4.y + r4.y;
    o.z = c.z * inv * g4.z + e4.z + r4.z;
    o.w = c.w * inv * g4.w + e4.w + r4.w;
    ((float4*)(xio + (size_t)n * GD))[lane] = o;
}

// ---------------------------------------------------------------------------
extern "C" void kernel_launch(void* const* d_in, const int* in_sizes, int n_in,
                              void* d_out, int out_size, void* d_ws, size_t ws_size,
                              hipStream_t stream)
{
    (void)in_sizes; (void)n_in; (void)out_size; (void)ws_size;
    const float* x     = (const float*)d_in[0];
    const int*   ei    = (const int*)  d_in[1];
    const float* Wl    = (const float*)d_in[2];
    const float* bl    = (const float*)d_in[3];
    const float* Wr    = (const float*)d_in[4];
    const float* br    = (const float*)d_in[5];
    const float* att   = (const float*)d_in[6];
    const float* bias  = (const float*)d_in[7];
    const float* gamma = (const float*)d_in[8];
    const float* beta  = (const float*)d_in[9];
    const int* srcA = ei;
    const int* dstA = ei + GE;

    // workspace layout (floats): agg | denom | mOrd | xl | xr | logit
    float* agg   = (float*)d_ws;
    float* denom = agg + (size_t)GN * GD;
    int*   mOrd  = (int*)(denom + (size_t)GN * GH);
    float* xl    = (float*)(mOrd + (size_t)GN * GH);
    float* xr    = xl + (size_t)GN * GD;
    float* logit = xr + (size_t)GN * GD;
    float* xcur  = (float*)d_out;   // node features; final layer output lands here

    hipMemcpyAsync(xcur, x, sizeof(float) * (size_t)GN * GD,
                   hipMemcpyDeviceToDevice, stream);

    const int nZero = GN * GD + GN * GH;      // agg + denom are contiguous
    const int nMax  = GN * GH;
    const int nRowTiles = (GN + 15) / 16;
    const int nGemmBlk  = (nRowTiles + BT - 1) / BT;

    for (int l = 0; l < GL; ++l) {
        const float* Wl_l = Wl   + (size_t)l * GIN * GD;
        const float* bl_l = bl   + (size_t)l * GD;
        const float* Wr_l = Wr   + (size_t)l * GIN * GD;
        const float* br_l = br   + (size_t)l * GD;
        const float* at_l = att  + (size_t)l * GH * GC;
        const float* bi_l = bias + (size_t)l * GD;
        const float* ga_l = gamma+ (size_t)l * GD;
        const float* be_l = beta + (size_t)l * GD;

        fill_u32<<<(nZero + 255) / 256, 256, 0, stream>>>((unsigned*)agg, 0u, nZero);
        fill_u32<<<(nMax + 255) / 256, 256, 0, stream>>>((unsigned*)mOrd,
                                                         0x80000000u, nMax);
        gemm_xl_xr<<<nGemmBlk, 256, 0, stream>>>(xcur, Wl_l, bl_l, Wr_l, br_l,
                                                 xl, xr);
        edge_logit<<<GE / 8, 256, 0, stream>>>(xl, xr, srcA, dstA, at_l,
                                               logit, mOrd);
        edge_exp<<<(GE * GH + 255) / 256, 256, 0, stream>>>(logit, dstA, mOrd,
                                                            denom);
        edge_aggr<<<GE / 8, 256, 0, stream>>>(logit, denom, xl, srcA, dstA, agg);
        node_fin<<<GN / 8, 256, 0, stream>>>(agg, bi_l, ga_l, be_l, xcur);
    }
}